// GameTheory_5025111736961
// MI455X (gfx1250) — compile-verified
//
#include <hip/hip_runtime.h>
#include <hip/hip_bf16.h>

typedef __attribute__((ext_vector_type(2))) float v2f;
typedef __attribute__((ext_vector_type(8))) float v8f;

#define NUM_M 1024
#define NUM_D 256
#define F_M   256
#define F_D   384
#define HDIM  128
#define PAIRS (NUM_M * NUM_D)

// ---------------------------------------------------------------------------
// Kernel: Out[R x 128] = E[R x F] @ W^T + bias   (V_WMMA_F32_16X16X4_F32)
// Block = 256 threads (8 waves). blockIdx.x = 16-row tile, wave = 16-col tile.
// A 16x4 f32 layout: lane L -> row L%16, x=K(half*2), y=K(half*2+1)
// B 4x16  f32 layout: lane L -> col L%16, x=K(half*2), y=K(half*2+1)
// C/D 16x16: VGPR r -> M = r + half*8, N = L%16
// ---------------------------------------------------------------------------
__global__ void proj_kernel(const float* __restrict__ E, const float* __restrict__ W,
                            const float* __restrict__ bias, float* __restrict__ Out,
                            int F) {
  const int wave = threadIdx.x >> 5;
  const int lane = threadIdx.x & 31;
  const int half = lane >> 4;
  const int r16  = lane & 15;
  const int rowBase = blockIdx.x * 16;
  const int nBase   = wave * 16;
  const float* aRow = E + (size_t)(rowBase + r16) * F + half * 2;
  const float* bRow = W + (size_t)(nBase   + r16) * F + half * 2;   // B = W^T
  v8f c = {0.f, 0.f, 0.f, 0.f, 0.f, 0.f, 0.f, 0.f};
  for (int k = 0; k < F; k += 4) {
    v2f a; a.x = aRow[k]; a.y = aRow[k + 1];
    v2f b; b.x = bRow[k]; b.y = bRow[k + 1];
    c = __builtin_amdgcn_wmma_f32_16x16x4_f32(false, a, false, b, (short)0, c,
                                              false, false);
  }
  const float bv = bias[nBase + r16];
#pragma unroll
  for (int r = 0; r < 8; ++r) {
    int M = r + half * 8;
    Out[(size_t)(rowBase + M) * HDIM + nBase + r16] = c[r] + bv;
  }
}

// ---------------------------------------------------------------------------
// Kernel: rinv[row] = rsqrt(sum_h X[row][h]^2)   one wave per row
// ---------------------------------------------------------------------------
__global__ void rnorm_kernel(const float* __restrict__ X, float* __restrict__ rinv,
                             int R) {
  const int row  = blockIdx.x * (blockDim.x >> 5) + (threadIdx.x >> 5);
  const int lane = threadIdx.x & 31;
  if (row >= R) return;
  float s = 0.f;
  for (int h = lane; h < HDIM; h += 32) {
    float v = X[(size_t)row * HDIM + h];
    s += v * v;
  }
#pragma unroll
  for (int off = 16; off; off >>= 1) s += __shfl_xor(s, off, 32);
  if (lane == 0) rinv[row] = rsqrtf(s);
}

// ---------------------------------------------------------------------------
// Kernel: C[1024 x 256] = (Mp*rM) @ (Dp*rD)^T    (WMMA f32 16x16x4)
// 64x16 = 1024 tiles, 8 tiles per block.
// ---------------------------------------------------------------------------
__global__ void cosine_kernel(const float* __restrict__ Mp, const float* __restrict__ Dp,
                              const float* __restrict__ mRn, const float* __restrict__ dRn,
                              float* __restrict__ C) {
  const int wave = threadIdx.x >> 5;
  const int lane = threadIdx.x & 31;
  const int half = lane >> 4;
  const int r16  = lane & 15;
  const int tile = blockIdx.x * 8 + wave;  // 0..1023
  const int tM = tile >> 4;                // 0..63
  const int tN = tile & 15;                // 0..15
  const int row = tM * 16 + r16;
  const int col = tN * 16 + r16;
  const float ra = mRn[row];
  const float rb = dRn[col];
  const float* aRow = Mp + (size_t)row * HDIM + half * 2;
  const float* bRow = Dp + (size_t)col * HDIM + half * 2;
  v8f c = {0.f, 0.f, 0.f, 0.f, 0.f, 0.f, 0.f, 0.f};
  for (int k = 0; k < HDIM; k += 4) {
    v2f a; a.x = aRow[k] * ra; a.y = aRow[k + 1] * ra;
    v2f b; b.x = bRow[k] * rb; b.y = bRow[k + 1] * rb;
    c = __builtin_amdgcn_wmma_f32_16x16x4_f32(false, a, false, b, (short)0, c,
                                              false, false);
  }
#pragma unroll
  for (int r = 0; r < 8; ++r) {
    int M = tM * 16 + r + half * 8;
    C[(size_t)M * NUM_D + tN * 16 + r16] = c[r];
  }
}

// ---------------------------------------------------------------------------
// argmax helpers: first-occurrence max (matches jnp.argmax)
// ---------------------------------------------------------------------------
__global__ void init_kernel(float* acc) { acc[0] = 0.f; acc[1] = 0.f; }

// one wave per miRNA row: best column j in [0,256); bm_row[mi]=miRNA_index[j];
// accumulate ||Mp[mi]-Mp[bm]||^2 into acc[0]
__global__ void argmax_rows_kernel(const float* __restrict__ C,
                                   const int* __restrict__ m_idx,
                                   const float* __restrict__ Mp,
                                   int* __restrict__ bm_row, float* acc) {
  const int row  = blockIdx.x * (blockDim.x >> 5) + (threadIdx.x >> 5);
  const int lane = threadIdx.x & 31;
  float best = -__builtin_inff();
  int bidx = 0x7fffffff;
  for (int j = lane; j < NUM_D; j += 32) {
    float v = C[(size_t)row * NUM_D + j];
    if (v > best) { best = v; bidx = j; }
  }
#pragma unroll
  for (int off = 16; off; off >>= 1) {
    float ov = __shfl_xor(best, off, 32);
    int   oi = __shfl_xor(bidx, off, 32);
    if (ov > best || (ov == best && oi < bidx)) { best = ov; bidx = oi; }
  }
  const int bm = m_idx[bidx];            // pair row bidx<256 -> its miRNA index
  if (lane == 0) bm_row[row] = bm;
  float s = 0.f;
  for (int h = lane; h < HDIM; h += 32) {
    float df = Mp[(size_t)row * HDIM + h] - Mp[(size_t)bm * HDIM + h];
    s += df * df;
  }
#pragma unroll
  for (int off = 16; off; off >>= 1) s += __shfl_xor(s, off, 32);
  if (lane == 0) atomicAdd(&acc[0], s);
}

// one wave per disease column: best row i in [0,1024); bd_row[di]=disease_index[i]
__global__ void argmax_cols_kernel(const float* __restrict__ C,
                                   const int* __restrict__ d_idx,
                                   const float* __restrict__ Dp,
                                   int* __restrict__ bd_row, float* acc) {
  const int col  = blockIdx.x * (blockDim.x >> 5) + (threadIdx.x >> 5);
  const int lane = threadIdx.x & 31;
  float best = -__builtin_inff();
  int bidx = 0x7fffffff;
  for (int i = lane; i < NUM_M; i += 32) {
    float v = C[(size_t)i * NUM_D + col];
    if (v > best) { best = v; bidx = i; }
  }
#pragma unroll
  for (int off = 16; off; off >>= 1) {
    float ov = __shfl_xor(best, off, 32);
    int   oi = __shfl_xor(bidx, off, 32);
    if (ov > best || (ov == best && oi < bidx)) { best = ov; bidx = oi; }
  }
  const int bd = d_idx[bidx];            // pair row bidx<1024 -> its disease index
  if (lane == 0) bd_row[col] = bd;
  float s = 0.f;
  for (int h = lane; h < HDIM; h += 32) {
    float df = Dp[(size_t)col * HDIM + h] - Dp[(size_t)bd * HDIM + h];
    s += df * df;
  }
#pragma unroll
  for (int off = 16; off; off >>= 1) s += __shfl_xor(s, off, 32);
  if (lane == 0) atomicAdd(&acc[1], s);
}

__global__ void finalize_kernel(const float* acc, float* out0) {
  // each mi repeats NUM_D times, each di repeats NUM_M times
  float mean_m = acc[0] / ((float)NUM_M * (float)HDIM);
  float mean_d = acc[1] / ((float)NUM_D * (float)HDIM);
  out0[0] = 0.5f * (mean_m + mean_d);
}

// ---------------------------------------------------------------------------
// Streaming kernel: one wave per pair. Gathers L2-resident strategy rows,
// writes 2x128 f32 per pair with non-temporal stores (268 MB total -> HBM bound)
// d_out+1 is only 4B-aligned, so use coalesced b32 stores.
// ---------------------------------------------------------------------------
__global__ void scatter_kernel(const int* __restrict__ m_idx, const int* __restrict__ d_idx,
                               const int* __restrict__ bm_row, const int* __restrict__ bd_row,
                               const float* __restrict__ Mp, const float* __restrict__ Dp,
                               float* __restrict__ out_m, float* __restrict__ out_d) {
  const int p    = blockIdx.x * (blockDim.x >> 5) + (threadIdx.x >> 5);
  const int lane = threadIdx.x & 31;
  const int mi = m_idx[p];
  const int di = d_idx[p];
  const int bm = bm_row[mi];
  const int bd = bd_row[di];
  const float* ms = Mp + (size_t)bm * HDIM;
  const float* ds = Dp + (size_t)bd * HDIM;
  const size_t op = (size_t)p * HDIM;
#pragma unroll
  for (int j = 0; j < 4; ++j) {
    __builtin_nontemporal_store(ms[lane + 32 * j], &out_m[op + lane + 32 * j]);
    __builtin_nontemporal_store(ds[lane + 32 * j], &out_d[op + lane + 32 * j]);
  }
}

// ---------------------------------------------------------------------------
extern "C" void kernel_launch(void* const* d_in, const int* in_sizes, int n_in,
                              void* d_out, int out_size, void* d_ws, size_t ws_size,
                              hipStream_t stream) {
  (void)in_sizes; (void)n_in; (void)out_size; (void)ws_size;
  const float* mE   = (const float*)d_in[0];   // [1024,256]
  const float* dE   = (const float*)d_in[1];   // [256,384]
  const float* Wm   = (const float*)d_in[2];   // [128,256]
  const float* bm   = (const float*)d_in[3];   // [128]
  const float* Wd   = (const float*)d_in[4];   // [128,384]
  const float* bd   = (const float*)d_in[5];   // [128]
  const int*  m_idx = (const int*)d_in[6];     // [P]
  const int*  d_idx = (const int*)d_in[7];     // [P]
  float* out = (float*)d_out;                  // [1 + P*128 + P*128]

  // workspace layout (< 2 MB)
  char* w = (char*)d_ws;
  float* Mproj  = (float*)(w);                          // 512 KB
  float* Dproj  = (float*)(w + (512u << 10));           // 128 KB
  float* Mrn    = (float*)(w + (640u << 10));           //   4 KB
  float* Drn    = (float*)(w + (644u << 10));           //   1 KB
  float* acc    = (float*)(w + (645u << 10));           //   8 B
  float* Cmat   = (float*)(w + (646u << 10));           //   1 MB
  int*   bm_row = (int*)  (w + (1670u << 10));          //   4 KB
  int*   bd_row = (int*)  (w + (1674u << 10));          //   1 KB

  init_kernel<<<1, 1, 0, stream>>>(acc);
  proj_kernel<<<NUM_M / 16, 256, 0, stream>>>(mE, Wm, bm, Mproj, F_M);
  proj_kernel<<<NUM_D / 16, 256, 0, stream>>>(dE, Wd, bd, Dproj, F_D);
  rnorm_kernel<<<NUM_M / 8, 256, 0, stream>>>(Mproj, Mrn, NUM_M);
  rnorm_kernel<<<NUM_D / 8, 256, 0, stream>>>(Dproj, Drn, NUM_D);
  cosine_kernel<<<(NUM_M / 16) * (NUM_D / 16) / 8, 256, 0, stream>>>(Mproj, Dproj,
                                                                     Mrn, Drn, Cmat);
  argmax_rows_kernel<<<NUM_M / 8, 256, 0, stream>>>(Cmat, m_idx, Mproj, bm_row, acc);
  argmax_cols_kernel<<<NUM_D / 8, 256, 0, stream>>>(Cmat, d_idx, Dproj, bd_row, acc);
  finalize_kernel<<<1, 1, 0, stream>>>(acc, out);

  float* out_m = out + 1;
  float* out_d = out + 1 + (size_t)PAIRS * HDIM;
  scatter_kernel<<<PAIRS / 8, 256, 0, stream>>>(m_idx, d_idx, bm_row, bd_row,
                                                Mproj, Dproj, out_m, out_d);
}